// GraphConvGRUCell_25271587570212
// MI455X (gfx1250) — compile-verified
//
#include <hip/hip_runtime.h>
#include <math.h>

#define NN  10000   // nodes
#define BB  8       // batch
#define HH  128     // hidden
#define DIN 256     // input dim

typedef __attribute__((ext_vector_type(2))) float v2f;
typedef __attribute__((ext_vector_type(8))) float v8f;

__device__ __forceinline__ float sigmoidf_(float x) {
    return 1.0f / (1.0f + __expf(-x));
}

// ---------------------------------------------------------------- zero scratch
__global__ void zero_ws_kernel(float* __restrict__ ws, int n) {
    int i = blockIdx.x * blockDim.x + threadIdx.x;
    if (i < n) ws[i] = 0.0f;
}

// ---------------------------------------------------------------- degrees
__global__ void degree_kernel(const int* __restrict__ src, const int* __restrict__ dst,
                              float* __restrict__ outdeg, float* __restrict__ indeg, int E) {
    int e = blockIdx.x * blockDim.x + threadIdx.x;
    if (e >= E) return;
    atomicAdd(&outdeg[src[e]], 1.0f);
    atomicAdd(&indeg[dst[e]], 1.0f);
}

// in-place deg -> rsqrt(deg) (0 if deg==0)
__global__ void norm_kernel(float* __restrict__ deg_out, float* __restrict__ deg_in) {
    int i = blockIdx.x * blockDim.x + threadIdx.x;
    if (i >= NN) return;
    float o = deg_out[i]; deg_out[i] = (o > 0.0f) ? rsqrtf(o) : 0.0f;
    float d = deg_in[i];  deg_in[i]  = (d > 0.0f) ? rsqrtf(d) : 0.0f;
}

// ---------------------------------------------------------------- x @ {w_r,w_z,w_h} + b   (tiny: 3*8*128 outputs)
__global__ void xgemm_kernel(const float* __restrict__ x,
                             const float* __restrict__ wr, const float* __restrict__ br,
                             const float* __restrict__ wz, const float* __restrict__ bz,
                             const float* __restrict__ wh, const float* __restrict__ bh,
                             float* __restrict__ xout /* [3][B][H] */) {
    int i = blockIdx.x * blockDim.x + threadIdx.x;
    if (i >= 3 * BB * HH) return;
    int which = i / (BB * HH);
    int rem   = i % (BB * HH);
    int b = rem / HH, c = rem % HH;
    const float* W  = (which == 0) ? wr : ((which == 1) ? wz : wh);
    const float* Bv = (which == 0) ? br : ((which == 1) ? bz : bh);
    float acc = Bv[c];
    const float* xp = x + b * DIN;
    for (int k = 0; k < DIN; ++k) acc += xp[k] * W[k * HH + c];
    xout[i] = acc;
}

// ---------------------------------------------------------------- edge scatter (L2-resident atomics)
// agg[b, dst[e], :] += h_prev[b, src[e], :] * norm_out[src[e]]
__global__ void scatter_kernel(const float* __restrict__ h_prev,
                               const int* __restrict__ src, const int* __restrict__ dst,
                               const float* __restrict__ norm_out,
                               float* __restrict__ agg, int E) {
    long long idx = (long long)blockIdx.x * blockDim.x + threadIdx.x;  // E * (H/4)
    if (idx >= (long long)E * (HH / 4)) return;
    int e  = (int)(idx >> 5);
    int c4 = (int)(idx & 31) << 2;
    int s = src[e], d = dst[e];
    float ns = norm_out[s];
    for (int b = 0; b < BB; ++b) {
        const float4 v = *(const float4*)(h_prev + ((long long)b * NN + s) * HH + c4);
        float* ap = agg + ((long long)b * NN + d) * HH + c4;
        atomicAdd(ap + 0, v.x * ns);
        atomicAdd(ap + 1, v.y * ns);
        atomicAdd(ap + 2, v.z * ns);
        atomicAdd(ap + 3, v.w * ns);
    }
}

// ---------------------------------------------------------------- fused WMMA GEMM + GRU epilogue
// h_conv = (agg * norm_in) @ gcn_w + gcn_b     via V_WMMA_F32_16X16X4_F32 (full fp32)
// out    = (1-z)*h_prev + z*tanh(xh + r*h_conv),  r/z = sigmoid(xr/xz + h_conv)
// One block = one 16-row tile of M = B*N (N divisible by 16 -> tile never straddles batches).
// 8 waves per block, wave w owns column tile [16w, 16w+16). A-tile staged in LDS once.
__global__ void __launch_bounds__(256)
wmma_gru_kernel(const float* __restrict__ agg,   const float* __restrict__ norm_in,
                const float* __restrict__ gcn_w, const float* __restrict__ gcn_b,
                const float* __restrict__ xrzh,  /* [3][B][H] */
                const float* __restrict__ h_prev,
                float* __restrict__ out) {
    __shared__ float ldsA[16 * HH];

    const int row0  = blockIdx.x * 16;
    const int batch = row0 / NN;            // whole tile in one batch (NN % 16 == 0)

    // Stage A tile (scaled by norm_in) into LDS: 2048 floats / 256 threads.
    for (int i = threadIdx.x; i < 16 * HH; i += 256) {
        int rr = i >> 7, cc = i & (HH - 1);
        int row  = row0 + rr;
        int node = row - batch * NN;
        ldsA[i] = agg[(long long)row * HH + cc] * norm_in[node];
    }
    __syncthreads();

    const int lane = threadIdx.x & 31;
    const int wid  = threadIdx.x >> 5;      // 0..7 column tile
    const int col0 = wid * 16;
    const int half = lane >> 4;             // 0: lanes 0-15, 1: lanes 16-31
    const int m    = lane & 15;             // A row / B,C,D column within tile

    v8f c = {};
    // K loop: 32 x V_WMMA_F32_16X16X4_F32.
    // f32 A 16x4 layout: VGPR0 = K (k0 + 2*half), VGPR1 = K+1; B 4x16 mirrors it.
    for (int k0 = 0; k0 < HH; k0 += 4) {
        const int ka = k0 + 2 * half;
        v2f a = { ldsA[m * HH + ka], ldsA[m * HH + ka + 1] };
        v2f b = { gcn_w[(ka    ) * HH + col0 + m],
                  gcn_w[(ka + 1) * HH + col0 + m] };
        c = __builtin_amdgcn_wmma_f32_16x16x4_f32(
                /*neg_a=*/false, a, /*neg_b=*/false, b,
                /*c_mod=*/(short)0, c, /*reuse_a=*/false, /*reuse_b=*/false);
    }

    // GRU epilogue on the 16x16 f32 D tile: VGPR r -> M = r + 8*half, N = m.
    const int   col  = col0 + m;
    const float bias = gcn_b[col];
    const float xr = xrzh[0 * BB * HH + batch * HH + col];
    const float xz = xrzh[1 * BB * HH + batch * HH + col];
    const float xh = xrzh[2 * BB * HH + batch * HH + col];
#pragma unroll
    for (int r = 0; r < 8; ++r) {
        int   row = row0 + r + 8 * half;
        float hc  = c[r] + bias;
        float hp  = h_prev[(long long)row * HH + col];
        float rg  = sigmoidf_(xr + hc);
        float zg  = sigmoidf_(xz + hc);
        float ht  = tanhf(xh + rg * hc);
        out[(long long)row * HH + col] = (1.0f - zg) * hp + zg * ht;
    }
}

// ---------------------------------------------------------------- launch
extern "C" void kernel_launch(void* const* d_in, const int* in_sizes, int n_in,
                              void* d_out, int out_size, void* d_ws, size_t ws_size,
                              hipStream_t stream) {
    const float* x      = (const float*)d_in[0];
    const float* h_prev = (const float*)d_in[1];
    const int*   src    = (const int*)  d_in[2];
    const int*   dst    = (const int*)  d_in[3];
    const float* w_r    = (const float*)d_in[4];
    const float* b_r    = (const float*)d_in[5];
    const float* w_z    = (const float*)d_in[6];
    const float* b_z    = (const float*)d_in[7];
    const float* w_h    = (const float*)d_in[8];
    const float* b_h    = (const float*)d_in[9];
    const float* gcn_w  = (const float*)d_in[10];
    const float* gcn_b  = (const float*)d_in[11];
    float* out = (float*)d_out;
    const int E = in_sizes[2];

    // Workspace layout (floats): norm_out[N] | norm_in[N] | xrzh[3*B*H] | agg[B*N*H]
    float* ws       = (float*)d_ws;
    float* norm_out = ws;
    float* norm_in  = ws + NN;
    float* xrzh     = ws + 2 * NN;
    float* agg      = ws + 2 * NN + 3 * BB * HH;   // 16-float aligned offset

    const int n_zero = 2 * NN + 3 * BB * HH + BB * NN * HH;
    zero_ws_kernel<<<(n_zero + 255) / 256, 256, 0, stream>>>(ws, n_zero);

    degree_kernel<<<(E + 255) / 256, 256, 0, stream>>>(src, dst, norm_out, norm_in, E);
    norm_kernel<<<(NN + 255) / 256, 256, 0, stream>>>(norm_out, norm_in);

    xgemm_kernel<<<(3 * BB * HH + 255) / 256, 256, 0, stream>>>(
        x, w_r, b_r, w_z, b_z, w_h, b_h, xrzh);

    const long long sc = (long long)E * (HH / 4);
    scatter_kernel<<<(int)((sc + 255) / 256), 256, 0, stream>>>(
        h_prev, src, dst, norm_out, agg, E);

    wmma_gru_kernel<<<(BB * NN) / 16, 256, 0, stream>>>(
        agg, norm_in, gcn_w, gcn_b, xrzh, h_prev, out);
}